// LinearNotFirst_35940286333596
// MI455X (gfx1250) — compile-verified
//
#include <hip/hip_runtime.h>
#include <math.h>

#define BATCH 512
#define DIM   256
#define UNITS 256
#define NK4   (DIM / 4)            // 64 K-steps of 4
#define NT    (UNITS / 16)         // 16 column tiles of w_mu
#define TF_BYTES    (NK4 * 8 * 256)        // Tf[k4][8 jtiles][32 lanes]{v2f} = 128 KB
#define STAGE_PITCH 68                     // 64 floats + 4 pad (TDM pad_enable)
#define STAGE_BYTES (DIM * STAGE_PITCH * 4) // 69632 B per chunk buffer
#define STAGE_OFF   TF_BYTES
#define LDS_BYTES   (TF_BYTES + 2 * STAGE_BYTES) // 270336 B (<320 KB WGP LDS)

typedef __attribute__((ext_vector_type(2))) float    v2f;
typedef __attribute__((ext_vector_type(8))) float    v8f;
typedef __attribute__((ext_vector_type(4))) unsigned u32x4;
typedef __attribute__((ext_vector_type(8))) int      i32x8;
typedef __attribute__((ext_vector_type(4))) int      i32x4;

#if __has_builtin(__builtin_amdgcn_tensor_load_to_lds)
#define HAVE_TDM 1
#else
#define HAVE_TDM 0
#endif

__device__ __forceinline__ v8f wmma_f32_16x16x4(v2f a, v2f b, v8f c) {
    return __builtin_amdgcn_wmma_f32_16x16x4_f32(false, a, false, b, (short)0, c,
                                                 false, false);
}

#if HAVE_TDM
// TDM: DMA a 256-row x 64-float column chunk of Sigma_b into LDS at lds_off,
// padding 4 dwords after every 64 (row) so the staged pitch is 68 floats.
// D# fields per CDNA5 ISA 8.3/8.4 (2D tile, data_size=4B, type=2).
// This toolchain's builtin takes 6 args (g0, g1, g2, g3, g4, cpol).
__device__ __forceinline__ void tdm_load_chunk(const float* gbase, unsigned lds_off) {
    const unsigned long long ga = (unsigned long long)(size_t)gbase;
    u32x4 g0;
    g0[0] = 1u;                                        // count=1 (valid user D#)
    g0[1] = lds_off;                                   // LDS byte address
    g0[2] = (unsigned)(ga & 0xffffffffu);              // global_addr[31:0]
    g0[3] = (unsigned)((ga >> 32) & 0x01ffffffu)       // global_addr[56:32]
          | (2u << 30);                                // type=2 ("image")
    i32x8 g1;
    g1[0] = (2 << 16)                                  // data_size=4B
          | (1 << 20)                                  // pad_enable
          | (5 << 22)                                  // pad_interval: every 64 dw
          | (3 << 25);                                 // pad_amount: 4 dwords
    g1[1] = (int)(256u << 16);                         // tensor_dim0 = 256 (lo16)
    g1[2] = (int)(256u << 16);                         // dim0 hi | tensor_dim1 = 256
    g1[3] = (int)(64u << 16);                          // dim1 hi | tile_dim0 = 64
    g1[4] = 256;                                       // tile_dim1 = 256 rows
    g1[5] = 256;                                       // tensor_dim0_stride = 256 elems
    g1[6] = 0;
    g1[7] = 0;
    const i32x4 z4 = {0, 0, 0, 0};                     // 2D: groups 2/3 unused
    const i32x8 z8 = {0, 0, 0, 0, 0, 0, 0, 0};
    __builtin_amdgcn_tensor_load_to_lds(g0, g1, z4, z4, z8, 0);
}
#endif

// Pack w_mu into WMMA fragment order (one coalesced 256B b64 wave-load / frag).
__global__ void pack_kernel(const float* __restrict__ w_mu, v2f* __restrict__ F) {
    const int k4 = blockIdx.x;
    for (int e = threadIdx.x; e < NT * 32; e += blockDim.x) {
        const int tile = e >> 5;
        const int lane = e & 31;
        const int half = lane >> 4, l15 = lane & 15;
        const int col = tile * 16 + l15;
        const int row = k4 * 4 + half * 2;
        v2f v;
        v.x = w_mu[row * UNITS + col];
        v.y = w_mu[(row + 1) * UNITS + col];
        F[((size_t)k4 * NT + tile) * 32 + lane] = v;
    }
}

// mu_out = mu_in @ w_mu ; s_ws[b] = x.x + tr(Sigma_b) ; sp_ws = softplus(w_sigma)
__global__ void prep_kernel(const float* __restrict__ mu_in,
                            const float* __restrict__ Sigma_in,
                            const float* __restrict__ w_mu,
                            const float* __restrict__ w_sigma,
                            float* __restrict__ mu_out,
                            float* __restrict__ s_ws,
                            float* __restrict__ sp_ws) {
    const int b = blockIdx.x;
    const int t = threadIdx.x;
    __shared__ float xrow[DIM];
    __shared__ float red[DIM];

    const float xt = mu_in[b * DIM + t];
    xrow[t] = xt;
    __syncthreads();

    float acc = 0.f;
#pragma unroll 8
    for (int d = 0; d < DIM; ++d)
        acc = fmaf(xrow[d], w_mu[d * UNITS + t], acc);
    mu_out[b * UNITS + t] = acc;

    red[t] = xt * xt + Sigma_in[(size_t)b * DIM * DIM + (size_t)t * (DIM + 1)];
    __syncthreads();
    for (int off = DIM / 2; off > 0; off >>= 1) {
        if (t < off) red[t] += red[t + off];
        __syncthreads();
    }
    if (t == 0) s_ws[b] = red[0];
    if (b == 0) sp_ws[t] = log1pf(expf(w_sigma[t]));
}

// Phase-1 accumulators -> fragment-ordered LDS (ds_store_b64, XOR bank skew).
__device__ __forceinline__ void store_tiles(char* lds, const v8f* acc, int it,
                                            int half, int l15) {
#pragma unroll
    for (int j = 0; j < 8; ++j)
#pragma unroll
        for (int i = 0; i < 4; ++i) {
            const int k4p = it * 4 + (i >> 1) + half * 2;
            const int lnP = (i & 1) * 16 + l15;
            const unsigned off =
                (unsigned)((k4p * 8 + j) * 256 + lnP * 8) ^ ((((unsigned)k4p >> 1) & 1u) << 7);
            v2f v;
            v.x = acc[j][2 * i];
            v.y = acc[j][2 * i + 1];
            *(v2f*)(lds + off) = v;
        }
}

// Fused epilogue: diagonal (Sigma_2+Sigma_3) add, NaN/Inf scrub, |diag|, store.
__device__ __forceinline__ void epilogue(float* __restrict__ outb, const v8f* acc,
                                         int ut, int jt0, int half, int l15,
                                         float sb, const float* __restrict__ sp) {
#pragma unroll
    for (int j = 0; j < 8; ++j)
#pragma unroll
        for (int r = 0; r < 8; ++r) {
            const int row = ut * 16 + r + half * 8;
            const int col = (jt0 + j) * 16 + l15;
            float x = acc[j][r];
            if (row == col) x += sb * sp[row];
            const unsigned bits = __float_as_uint(x);
            if ((bits & 0x7f800000u) == 0x7f800000u) x = 0.f;
            if (row == col) x = fabsf(x);
            outb[(size_t)row * UNITS + col] = x;
        }
}

// One block = (batch b, 128-wide column slice). TDM streams Sigma_b into LDS in
// double-buffered 64-column chunks while all 8 waves run f32 WMMA on the
// previous chunk; T stays in LDS in fragment order; phase 2 fuses the epilogue.
__global__ void __launch_bounds__(256, 1)
sigma_kernel(const float* __restrict__ Sigma_in,
             const v2f* __restrict__ F,
             const float* __restrict__ s_ws,
             const float* __restrict__ sp_ws,
             float* __restrict__ Sigma_out) {
    extern __shared__ char lds[];                 // [Tf 128K | stage0 68K | stage1 68K]
    const int b    = blockIdx.x >> 1;
    const int jt0  = (blockIdx.x & 1) * 8;
    const int lane = threadIdx.x & 31;
    const int wave = threadIdx.x >> 5;
    const int half = lane >> 4;
    const int l15  = lane & 15;
    const float* Sig = Sigma_in + (size_t)b * DIM * DIM;

    // ---------------- Phase 1: Tf = frag(Sigma_b @ w_mu[:, slice]) ---------
    const int it0 = wave, it1 = wave + 8;
    {
        v8f acc0[8] = {}, acc1[8] = {};
#if HAVE_TDM
        if (wave == 0) tdm_load_chunk(Sig, STAGE_OFF);
#endif
        for (int c = 0; c < 4; ++c) {
#if HAVE_TDM
            if (wave == 0) __builtin_amdgcn_s_wait_tensorcnt(0); // chunk c landed
            __syncthreads();                                     // publish + buffer-free
            if (c < 3 && wave == 0)                              // prefetch chunk c+1
                tdm_load_chunk(Sig + (size_t)(c + 1) * 64,
                               STAGE_OFF + (unsigned)(((c + 1) & 1) * STAGE_BYTES));
#else
            __syncthreads();                                     // buffer free
            for (int e = threadIdx.x; e < DIM * 16; e += 256) {  // fallback copy
                const int row = e >> 4, g = e & 15;
                const float4 v = *(const float4*)(Sig + (size_t)row * DIM + c * 64 + g * 4);
                *(float4*)(lds + STAGE_OFF + (size_t)((c & 1)) * STAGE_BYTES +
                           ((size_t)row * STAGE_PITCH + g * 4) * 4) = v;
            }
            __syncthreads();
#endif
            const char* sbuf = lds + STAGE_OFF + (size_t)(c & 1) * STAGE_BYTES;
            const char* a0p  = sbuf + ((size_t)(it0 * 16 + l15) * STAGE_PITCH + half * 2) * 4;
            const char* a1p  = sbuf + ((size_t)(it1 * 16 + l15) * STAGE_PITCH + half * 2) * 4;
            for (int kk = 0; kk < 64; kk += 4) {
                const v2f a0 = *(const v2f*)(a0p + kk * 4);      // ds_load_b64
                const v2f a1 = *(const v2f*)(a1p + kk * 4);
                const int k4 = c * 16 + (kk >> 2);
                const v2f* fb = F + ((size_t)k4 * NT + jt0) * 32 + lane;
#pragma unroll
                for (int j = 0; j < 8; ++j) {
                    const v2f bb = fb[(size_t)j * 32];           // coalesced 256B
                    acc0[j] = wmma_f32_16x16x4(a0, bb, acc0[j]);
                    acc1[j] = wmma_f32_16x16x4(a1, bb, acc1[j]);
                }
            }
        }
        store_tiles(lds, acc0, it0, half, l15);
        store_tiles(lds, acc1, it1, half, l15);
    }
    __syncthreads();

    // ---------------- Phase 2: Sigma_1 slice = w_mu^T @ T ------------------
    const float sb = s_ws[b];
    float* outb = Sigma_out + (size_t)b * UNITS * UNITS;
    const int ut0 = wave, ut1 = wave + 8;
    {
        v8f c0[8] = {}, c1[8] = {};
        for (int k4 = 0; k4 < NK4; ++k4) {
            const v2f a0 = F[((size_t)k4 * NT + ut0) * 32 + lane];
            const v2f a1 = F[((size_t)k4 * NT + ut1) * 32 + lane];
            const unsigned roff =
                (unsigned)(k4 * 8 * 256 + lane * 8) ^ ((((unsigned)k4 >> 1) & 1u) << 7);
#pragma unroll
            for (int j = 0; j < 8; ++j) {
                const v2f bb = *(const v2f*)(lds + (roff + j * 256)); // ds_load_b64
                c0[j] = wmma_f32_16x16x4(a0, bb, c0[j]);
                c1[j] = wmma_f32_16x16x4(a1, bb, c1[j]);
            }
        }
        epilogue(outb, c0, ut0, jt0, half, l15, sb, sp_ws);
        epilogue(outb, c1, ut1, jt0, half, l15, sb, sp_ws);
    }
}

extern "C" void kernel_launch(void* const* d_in, const int* in_sizes, int n_in,
                              void* d_out, int out_size, void* d_ws, size_t ws_size,
                              hipStream_t stream) {
    const float* mu_in    = (const float*)d_in[0];
    const float* Sigma_in = (const float*)d_in[1];
    const float* w_mu     = (const float*)d_in[2];
    const float* w_sigma  = (const float*)d_in[3];

    float* out       = (float*)d_out;
    float* mu_out    = out;                       // [512, 256]
    float* Sigma_out = out + BATCH * UNITS;       // [512, 256, 256]

    // Workspace: s[512] | sp[256] | F fragments (256 KB)
    float* s_ws  = (float*)d_ws;
    float* sp_ws = s_ws + BATCH;
    v2f*   F     = (v2f*)(sp_ws + UNITS);

    (void)hipFuncSetAttribute(reinterpret_cast<const void*>(sigma_kernel),
                              hipFuncAttributeMaxDynamicSharedMemorySize, LDS_BYTES);

    pack_kernel<<<NK4, 256, 0, stream>>>(w_mu, F);
    prep_kernel<<<BATCH, 256, 0, stream>>>(mu_in, Sigma_in, w_mu, w_sigma,
                                           mu_out, s_ws, sp_ws);
    sigma_kernel<<<BATCH * 2, 256, LDS_BYTES, stream>>>(Sigma_in, F, s_ws, sp_ws,
                                                        Sigma_out);
}